// OurModel_16149077033283
// MI455X (gfx1250) — compile-verified
//
#include <hip/hip_runtime.h>
#include <hip/hip_bf16.h>
#include <math.h>
#include <stdint.h>

// Problem constants (must match reference)
#define LL 4
#define BB 8
#define SS 2048
#define EE 2048
#define FF 8192
#define HH 8
#define DD 128
#define VV 32000

typedef __attribute__((ext_vector_type(2))) float v2f;
typedef __attribute__((ext_vector_type(8))) float v8f;

// ---------------------------------------------------------------------------
// gfx1250 async global->LDS staging (ASYNCcnt path), with safe fallback.
// Builtin signature (from hipcc diagnostic): pointer params are int* in the
// global / LDS address spaces.
// ---------------------------------------------------------------------------
#if __has_builtin(__builtin_amdgcn_global_load_async_to_lds_b32)
#define HAVE_ASYNC_LDS 1
typedef __attribute__((address_space(1))) int g_as_i32;
typedef __attribute__((address_space(3))) int l_as_i32;
#else
#define HAVE_ASYNC_LDS 0
#endif

__device__ __forceinline__ void stage_to_lds(const float* g, float* l) {
#if HAVE_ASYNC_LDS
  __builtin_amdgcn_global_load_async_to_lds_b32(
      (g_as_i32*)(uintptr_t)g,
      (l_as_i32*)(uintptr_t)l,
      0, 0);
#else
  *l = *g;
#endif
}

__device__ __forceinline__ void wait_async_done() {
#if HAVE_ASYNC_LDS
#if __has_builtin(__builtin_amdgcn_s_wait_asynccnt)
  __builtin_amdgcn_s_wait_asynccnt(0);
#else
  asm volatile("s_wait_asynccnt 0x0" ::: "memory");
#endif
#endif
}

// ---------------------------------------------------------------------------
// Embedding gather: h[b,e] = embedding[x[b], e]
// ---------------------------------------------------------------------------
__global__ __launch_bounds__(256) void embed_kernel(const int* __restrict__ x,
                                                    const float* __restrict__ emb,
                                                    float* __restrict__ h) {
  int i = blockIdx.x * 256 + threadIdx.x;          // over B*E
  if (i < BB * EE) {
    int b = i / EE, e = i % EE;
    h[i] = emb[(size_t)x[b] * EE + e];
  }
}

// ---------------------------------------------------------------------------
// LayerNorm (eps=1e-6) + posemb[pos] add.  One block per batch row.
// ---------------------------------------------------------------------------
__global__ __launch_bounds__(256) void ln_pos_kernel(const float* __restrict__ h,
                                                     const float* __restrict__ scale,
                                                     const float* __restrict__ bias,
                                                     const float* __restrict__ pe, // [S,E]
                                                     const int* __restrict__ posp,
                                                     float* __restrict__ out) {
  __shared__ float red[256];
  const int b = blockIdx.x, t = threadIdx.x;
  const int p = *posp;
  const float* hr = h + (size_t)b * EE;
  float s = 0.f;
  for (int e = t; e < EE; e += 256) s += hr[e];
  red[t] = s; __syncthreads();
  for (int w = 128; w > 0; w >>= 1) { if (t < w) red[t] += red[t + w]; __syncthreads(); }
  const float mu = red[0] * (1.0f / EE);
  __syncthreads();
  float v = 0.f;
  for (int e = t; e < EE; e += 256) { float d = hr[e] - mu; v += d * d; }
  red[t] = v; __syncthreads();
  for (int w = 128; w > 0; w >>= 1) { if (t < w) red[t] += red[t + w]; __syncthreads(); }
  const float rs = rsqrtf(red[0] * (1.0f / EE) + 1e-6f);
  const float* per = pe + (size_t)p * EE;
  for (int e = t; e < EE; e += 256)
    out[(size_t)b * EE + e] = (hr[e] - mu) * rs * scale[e] + bias[e] + per[e];
}

// ---------------------------------------------------------------------------
// Skinny GEMM via V_WMMA_F32_16X16X4_F32:  C[8,N] = A[8,K] x W
//   BT=false: W is [K,N] row-major.  BT=true: W is [N,K] row-major.
// Requirements: K % 256 == 0, N % 128 == 0 (true for all uses here).
// A is staged per-block into LDS in [8,256] chunks via async global->LDS
// (ASYNCcnt), bank-padded (+4 dwords/row) so the 8 rows hit distinct banks.
// M padded to 16 (rows 8..15 zeroed via mask; all loads in-bounds).
// One wave computes one 16x16 C tile; 8 waves (128 columns) per block.
// A-frag (32-bit 16x4): lane m=lane&15, kk=2*(lane>>4): {A[m][k+kk],A[m][k+kk+1]}
// C-frag: VGPR r -> row r (lanes 0-15) / r+8 (lanes 16-31), col = lane&15.
// ---------------------------------------------------------------------------
#define KC  256
#define KCP (KC + 4)

template <bool RELU, bool BT>
__global__ __launch_bounds__(256) void gemv_wmma_kernel(const float* __restrict__ A,
                                                        const float* __restrict__ W,
                                                        float* __restrict__ C,
                                                        int K, int N) {
  __shared__ float sa[8 * KCP];
  const int wid  = threadIdx.x >> 5;
  const int lane = threadIdx.x & 31;
  const int n0   = (blockIdx.x * 8 + wid) * 16;
  const int m    = lane & 15;
  const int kk   = (lane >> 4) << 1;            // 0 or 2
  const int n    = lane & 15;
  const float amask = (m < 8) ? 1.0f : 0.0f;    // zero the padding rows
  const float* arow = &sa[(m & 7) * KCP + kk];
  v8f acc = {};
  for (int k0 = 0; k0 < K; k0 += KC) {
    // Stage A chunk [8, KC] into LDS (direct-to-LDS async when available).
    for (int idx = threadIdx.x; idx < 8 * KC; idx += 256) {
      const int row = idx >> 8;                 // KC == 256
      const int col = idx & (KC - 1);
      stage_to_lds(A + (size_t)row * K + k0 + col, &sa[row * KCP + col]);
    }
    wait_async_done();
    __syncthreads();
#pragma unroll 8
    for (int k = 0; k < KC; k += 4) {
      v2f a, b;
      a.x = arow[k]     * amask;                // ds_read from staged A
      a.y = arow[k + 1] * amask;
      if (BT) {
        const float* Wp = W + (size_t)(n0 + n) * K + k0 + k + kk;
        __builtin_prefetch(Wp + KC, 0, 1);      // prefetch next K-chunk of W
        b.x = Wp[0];
        b.y = Wp[1];
      } else {
        const float* Wp = W + (size_t)(k0 + k + kk) * N + n0 + n;
        __builtin_prefetch(Wp + (size_t)KC * N, 0, 1);
        b.x = Wp[0];
        b.y = Wp[N];
      }
      acc = __builtin_amdgcn_wmma_f32_16x16x4_f32(false, a, false, b,
                                                  (short)0, acc, false, false);
    }
    __syncthreads();
  }
  const int mbase = (lane >> 4) * 8;            // rows 8..15 are padding
#pragma unroll
  for (int r = 0; r < 8; ++r) {
    int mm = mbase + r;
    if (mm < 8) {
      float v = acc[r];
      if (RELU) v = fmaxf(v, 0.0f);
      C[(size_t)mm * N + n0 + n] = v;
    }
  }
}

// ---------------------------------------------------------------------------
// Scatter k,v for current position into the (already copied) output caches.
// Layer base pointers passed in; layout within layer: [B,S,H,D].
// ---------------------------------------------------------------------------
__global__ __launch_bounds__(256) void scatter_kv_kernel(const float* __restrict__ kw,
                                                         const float* __restrict__ vw,
                                                         float* __restrict__ kc,
                                                         float* __restrict__ vc,
                                                         const int* __restrict__ posp) {
  int i = blockIdx.x * 256 + threadIdx.x;       // over B*H*D
  if (i >= BB * HH * DD) return;
  const int p  = *posp;
  const int b  = i / (HH * DD);
  const int hd = i % (HH * DD);
  size_t idx = ((size_t)b * SS + p) * (HH * DD) + hd;
  kc[idx] = kw[i];
  vc[idx] = vw[i];
}

// ---------------------------------------------------------------------------
// Decode attention, one block per (b,h).  Unscaled logits, mask s<=pos,
// softmax in LDS.  Pass 1: float4 K loads, bounded unroll (keeps VGPRs low).
// Pass 2: both 128-thread halves stream V (even/odd s), combined via LDS.
// ---------------------------------------------------------------------------
__global__ __launch_bounds__(256) void attention_kernel(const float* __restrict__ q,   // [B*H, D]
                                                        const float* __restrict__ kc,  // layer base [B,S,H,D]
                                                        const float* __restrict__ vc,
                                                        const int* __restrict__ posp,
                                                        float* __restrict__ o) {       // [B*H, D]
  __shared__ float sq[DD];
  __shared__ float sl[SS];
  __shared__ float red[256];
  const int p  = *posp;
  const int bh = blockIdx.x;
  const int b  = bh / HH, hh = bh % HH;
  const int t  = threadIdx.x;
  for (int d = t; d < DD; d += 256) sq[d] = q[(size_t)bh * DD + d];
  __syncthreads();
  const size_t base = (size_t)b * SS * HH * DD + (size_t)hh * DD;
  float lmax = -3.0e38f;
  for (int s = t; s <= p; s += 256) {
    const float4* kp = (const float4*)(kc + base + (size_t)s * (HH * DD));
    float dot = 0.f;
#pragma unroll 4
    for (int d4 = 0; d4 < DD / 4; ++d4) {
      const float4 kv = kp[d4];
      const float4 qv = *(const float4*)&sq[d4 * 4];
      dot += qv.x * kv.x + qv.y * kv.y + qv.z * kv.z + qv.w * kv.w;
    }
    sl[s] = dot;
    lmax = fmaxf(lmax, dot);
  }
  red[t] = lmax; __syncthreads();
  for (int w = 128; w > 0; w >>= 1) { if (t < w) red[t] = fmaxf(red[t], red[t + w]); __syncthreads(); }
  const float mx = red[0];
  __syncthreads();
  float lsum = 0.f;
  for (int s = t; s <= p; s += 256) { float w = __expf(sl[s] - mx); sl[s] = w; lsum += w; }
  red[t] = lsum; __syncthreads();
  for (int w = 128; w > 0; w >>= 1) { if (t < w) red[t] += red[t + w]; __syncthreads(); }
  const float inv = 1.0f / red[0];
  __syncthreads();
  // Pass 2: thread t handles d = t&127, group g = t>>7 takes s = g, g+2, ...
  const int d = t & (DD - 1);
  const int g = t >> 7;
  const float* vp = vc + base + d;
  float acc = 0.f;
  for (int s = g; s <= p; s += 2) acc += sl[s] * vp[(size_t)s * (HH * DD)];
  red[t] = acc; __syncthreads();
  if (t < DD) o[(size_t)bh * DD + t] = (red[t] + red[t + DD]) * inv;
}

// ---------------------------------------------------------------------------
extern "C" void kernel_launch(void* const* d_in, const int* in_sizes, int n_in,
                              void* d_out, int out_size, void* d_ws, size_t ws_size,
                              hipStream_t stream) {
  (void)in_sizes; (void)n_in; (void)out_size; (void)ws_size;
  const int*   x        = (const int*)  d_in[0];
  const int*   posp     = (const int*)  d_in[1];
  const float* kc_in    = (const float*)d_in[2];
  const float* vc_in    = (const float*)d_in[3];
  const float* emb      = (const float*)d_in[4];
  const float* ln_scale = (const float*)d_in[5];
  const float* ln_bias  = (const float*)d_in[6];
  const float* posemb   = (const float*)d_in[7];
  const float* w_ff     = (const float*)d_in[8];
  const float* w_ffout  = (const float*)d_in[9];
  const float* w_q      = (const float*)d_in[10];
  const float* w_k      = (const float*)d_in[11];
  const float* w_v      = (const float*)d_in[12];
  const float* w_o      = (const float*)d_in[13];

  float* out = (float*)d_out;
  const size_t cache_elems = (size_t)LL * BB * SS * HH * DD;
  float* kc_out = out + (size_t)BB * VV;
  float* vc_out = kc_out + cache_elems;

  // Output tuple includes the (updated) caches: copy, then patch pos entries.
  (void)hipMemcpyAsync(kc_out, kc_in, cache_elems * sizeof(float), hipMemcpyDeviceToDevice, stream);
  (void)hipMemcpyAsync(vc_out, vc_in, cache_elems * sizeof(float), hipMemcpyDeviceToDevice, stream);

  // Workspace carve-up (f32): h, h2 [B,E]; hf [B,F]; q,k,v,o [B,H*D]
  float* h  = (float*)d_ws;
  float* h2 = h  + (size_t)BB * EE;
  float* hf = h2 + (size_t)BB * EE;
  float* qw = hf + (size_t)BB * FF;
  float* kw = qw + (size_t)BB * HH * DD;
  float* vw = kw + (size_t)BB * HH * DD;
  float* ow = vw + (size_t)BB * HH * DD;

  embed_kernel<<<(BB * EE + 255) / 256, 256, 0, stream>>>(x, emb, h);

  const size_t layer_cache = (size_t)BB * SS * HH * DD;
  for (int i = 0; i < LL; ++i) {
    ln_pos_kernel<<<BB, 256, 0, stream>>>(h, ln_scale + (size_t)i * EE,
                                          ln_bias + (size_t)i * EE,
                                          posemb + (size_t)i * SS * EE, posp, h2);
    // h2[B,E] @ w_ff[E,F] -> relu -> hf
    gemv_wmma_kernel<true,  false><<<FF / 128, 256, 0, stream>>>(h2, w_ff + (size_t)i * EE * FF, hf, EE, FF);
    // hf[B,F] @ w_ffout[F,E] -> relu -> h
    gemv_wmma_kernel<true,  false><<<EE / 128, 256, 0, stream>>>(hf, w_ffout + (size_t)i * FF * EE, h, FF, EE);
    // q,k,v projections (E -> H*D)
    gemv_wmma_kernel<false, false><<<(HH * DD) / 128, 256, 0, stream>>>(h, w_q + (size_t)i * EE * HH * DD, qw, EE, HH * DD);
    gemv_wmma_kernel<false, false><<<(HH * DD) / 128, 256, 0, stream>>>(h, w_k + (size_t)i * EE * HH * DD, kw, EE, HH * DD);
    gemv_wmma_kernel<false, false><<<(HH * DD) / 128, 256, 0, stream>>>(h, w_v + (size_t)i * EE * HH * DD, vw, EE, HH * DD);
    // write current-token k,v into output caches
    scatter_kv_kernel<<<(BB * HH * DD + 255) / 256, 256, 0, stream>>>(
        kw, vw, kc_out + (size_t)i * layer_cache, vc_out + (size_t)i * layer_cache, posp);
    // attention over cache (includes just-written pos entry)
    attention_kernel<<<BB * HH, 256, 0, stream>>>(
        qw, kc_out + (size_t)i * layer_cache, vc_out + (size_t)i * layer_cache, posp, ow);
    // o[B,H*D] @ w_o[H*D,E] -> h
    gemv_wmma_kernel<false, false><<<EE / 128, 256, 0, stream>>>(ow, w_o + (size_t)i * HH * DD * EE, h, HH * DD, EE);
  }

  // Final logits: h[B,E] @ embedding^T, embedding stored [V,E] => BT path
  gemv_wmma_kernel<false, true><<<VV / 128, 256, 0, stream>>>(h, emb, out, EE, VV);
}